// LambdaNDCGLoss_54348516163729
// MI455X (gfx1250) — compile-verified
//
#include <hip/hip_runtime.h>
#include <hip/hip_bf16.h>

// Problem shape fixed by the reference
#define NCOLS 6000
#define NVEC  (NCOLS / 4)           // 1500 float4/int4 vectors per row
#define BLOCK 256
#define NWAVES (BLOCK / 32)
#define NCHUNK ((NCOLS * 4) / 16)   // 1500 b128 chunks per fp32 row
#define BIGF 1e30f

// 16-byte vector type matching the async builtin's parameter type exactly
typedef int v4i_t __attribute__((vector_size(16)));

// ---- CDNA5 async global->LDS copy (ASYNCcnt path) -------------------------
__device__ __forceinline__ void cp16_async(void* lds, const void* gbl) {
#if defined(__HIP_DEVICE_COMPILE__) && __has_builtin(__builtin_amdgcn_global_load_async_to_lds_b128)
    __builtin_amdgcn_global_load_async_to_lds_b128(
        (__attribute__((address_space(1))) v4i_t*)(gbl),
        (__attribute__((address_space(3))) v4i_t*)(lds),
        /*imm offset=*/0, /*cpol=*/0);
#else
    *(float4*)lds = *(const float4*)gbl;  // fallback: sync copy through VGPRs
#endif
}

__device__ __forceinline__ void wait_async0() {
#if defined(__HIP_DEVICE_COMPILE__)
#if __has_builtin(__builtin_amdgcn_s_wait_asynccnt)
    __builtin_amdgcn_s_wait_asynccnt(0);
#else
    asm volatile("s_wait_asynccnt 0" ::: "memory");
#endif
#endif
}

// ---- Kernel 1: one workgroup per row; single HBM pass via LDS staging -----
__global__ __launch_bounds__(BLOCK) void kl_row_kernel(
    const float* __restrict__ y_pred,
    const float* __restrict__ y_true,
    const int*   __restrict__ masks,
    float* __restrict__ row_out)
{
    extern __shared__ char smem[];
    float* syp = (float*)smem;            // NCOLS floats (16B aligned)
    float* syt = syp + NCOLS;             // offset 24000 (16B aligned)
    int*   smk = (int*)(syt + NCOLS);     // offset 48000 (16B aligned)

    __shared__ float warp_f[NWAVES][3];
    __shared__ int   warp_i[NWAVES];
    __shared__ float bc[4];

    const int row  = blockIdx.x;
    const int tid  = threadIdx.x;
    const int lane = tid & 31;
    const int wid  = tid >> 5;

    const float* gyp = y_pred + (size_t)row * NCOLS;
    const float* gyt = y_true + (size_t)row * NCOLS;
    const int*   gmk = masks  + (size_t)row * NCOLS;

    // Stage the whole row into LDS with async b128 copies (rows are 24000 B, 16B aligned)
    for (int c = tid; c < NCHUNK; c += BLOCK) {
        cp16_async(syp + c * 4, gyp + c * 4);
        cp16_async(syt + c * 4, gyt + c * 4);
        cp16_async(smk + c * 4, gmk + c * 4);
    }
    wait_async0();        // my wave's async writes to LDS are done
    __syncthreads();      // all waves' writes visible

    const float4* vyt = (const float4*)syt;
    const float4* vyp = (const float4*)syp;
    const int4*   vmk = (const int4*)smk;

    // ---- Pass 1 (from LDS, b128 reads): n, masked min/max(y_true), masked max(y_pred)
    int   cnt  = 0;
    float tmin =  BIGF, tmax = -BIGF, pmax = -BIGF;
    for (int i = tid; i < NVEC; i += BLOCK) {
        int4   m4 = vmk[i];
        float4 t4 = vyt[i];
        float4 p4 = vyp[i];
        if (m4.x > 0) { cnt++; tmin = fminf(tmin, t4.x); tmax = fmaxf(tmax, t4.x); pmax = fmaxf(pmax, p4.x); }
        if (m4.y > 0) { cnt++; tmin = fminf(tmin, t4.y); tmax = fmaxf(tmax, t4.y); pmax = fmaxf(pmax, p4.y); }
        if (m4.z > 0) { cnt++; tmin = fminf(tmin, t4.z); tmax = fmaxf(tmax, t4.z); pmax = fmaxf(pmax, p4.z); }
        if (m4.w > 0) { cnt++; tmin = fminf(tmin, t4.w); tmax = fmaxf(tmax, t4.w); pmax = fmaxf(pmax, p4.w); }
    }
    #pragma unroll
    for (int o = 16; o > 0; o >>= 1) {
        cnt  += __shfl_xor(cnt, o, 32);
        tmin  = fminf(tmin, __shfl_xor(tmin, o, 32));
        tmax  = fmaxf(tmax, __shfl_xor(tmax, o, 32));
        pmax  = fmaxf(pmax, __shfl_xor(pmax, o, 32));
    }
    if (lane == 0) {
        warp_i[wid]    = cnt;
        warp_f[wid][0] = tmin;
        warp_f[wid][1] = tmax;
        warp_f[wid][2] = pmax;
    }
    __syncthreads();
    if (tid == 0) {
        int c = 0; float a = BIGF, b = -BIGF, p = -BIGF;
        #pragma unroll
        for (int w = 0; w < NWAVES; ++w) {
            c += warp_i[w];
            a  = fminf(a, warp_f[w][0]);
            b  = fmaxf(b, warp_f[w][1]);
            p  = fmaxf(p, warp_f[w][2]);
        }
        bc[0] = (float)c; bc[1] = a; bc[2] = b; bc[3] = p;
    }
    __syncthreads();

    const float n = bc[0];
    tmin = bc[1]; tmax = bc[2]; pmax = bc[3];

    if (n < 2.0f) {                       // invalid row contributes 0
        if (tid == 0) row_out[row] = 0.0f;
        return;
    }

    const float rng   = tmax - tmin;
    const bool  donrm = rng > 0.0f;
    const float inv   = 1.0f / (rng + 1e-12f);
    // Analytic masked max of true_norm: value at the t_max element
    const float mt    = donrm ? rng * inv : tmax;

    // ---- Pass 2 (from LDS, b128 reads): exp-sums for both softmaxes + cross term
    float sA = 0.0f, sT = 0.0f, sP = 0.0f;
    for (int i = tid; i < NVEC; i += BLOCK) {
        int4   m4 = vmk[i];
        float4 t4 = vyt[i];
        float4 p4 = vyp[i];
        #pragma unroll
        for (int k = 0; k < 4; ++k) {
            int   mk = (k == 0) ? m4.x : (k == 1) ? m4.y : (k == 2) ? m4.z : m4.w;
            float t  = (k == 0) ? t4.x : (k == 1) ? t4.y : (k == 2) ? t4.z : t4.w;
            float p  = (k == 0) ? p4.x : (k == 1) ? p4.y : (k == 2) ? p4.z : p4.w;
            if (mk > 0) {
                float tn = donrm ? (t - tmin) * inv : t;
                float dt = tn - mt;
                float dp = p - pmax;
                float et = __expf(dt);
                sT += et;
                sA += et * (dt - dp);
                sP += __expf(dp);
            }
        }
    }
    #pragma unroll
    for (int o = 16; o > 0; o >>= 1) {
        sA += __shfl_xor(sA, o, 32);
        sT += __shfl_xor(sT, o, 32);
        sP += __shfl_xor(sP, o, 32);
    }
    if (lane == 0) {
        warp_f[wid][0] = sA;
        warp_f[wid][1] = sT;
        warp_f[wid][2] = sP;
    }
    __syncthreads();
    if (tid == 0) {
        float A = 0.0f, T = 0.0f, P = 0.0f;
        #pragma unroll
        for (int w = 0; w < NWAVES; ++w) {
            A += warp_f[w][0]; T += warp_f[w][1]; P += warp_f[w][2];
        }
        // KL = A/S_t + log(S_p) - log(S_t), then batchmean over n
        float kl = A / T + __logf(P) - __logf(T);
        row_out[row] = kl / n;
    }
}

// ---- Kernel 2: deterministic fixed-order reduction of 4096 row losses ----
__global__ __launch_bounds__(BLOCK) void kl_final_sum(
    const float* __restrict__ row_out, float* __restrict__ out, int B)
{
    __shared__ float sh[BLOCK];
    float s = 0.0f;
    for (int i = threadIdx.x; i < B; i += BLOCK) s += row_out[i];
    sh[threadIdx.x] = s;
    __syncthreads();
    #pragma unroll
    for (int w = BLOCK / 2; w > 0; w >>= 1) {
        if (threadIdx.x < w) sh[threadIdx.x] += sh[threadIdx.x + w];
        __syncthreads();
    }
    if (threadIdx.x == 0) out[0] = sh[0] / (float)B;
}

extern "C" void kernel_launch(void* const* d_in, const int* in_sizes, int n_in,
                              void* d_out, int out_size, void* d_ws, size_t ws_size,
                              hipStream_t stream) {
    const float* y_pred = (const float*)d_in[0];
    const float* y_true = (const float*)d_in[1];
    const int*   masks  = (const int*)d_in[2];
    float*       out    = (float*)d_out;
    float*       row_ws = (float*)d_ws;

    const int total = in_sizes[0];
    const int B     = total / NCOLS;     // 4096 for the reference shapes

    const size_t shbytes = (size_t)NCOLS * sizeof(float) * 2 + (size_t)NCOLS * sizeof(int); // 72000 B

    kl_row_kernel<<<B, BLOCK, shbytes, stream>>>(y_pred, y_true, masks, row_ws);
    kl_final_sum<<<1, BLOCK, 0, stream>>>(row_ws, out, B);
}